// CustomEmbedding_83391085019199
// MI455X (gfx1250) — compile-verified
//
#include <hip/hip_runtime.h>
#include <stdint.h>

// Embedding gather: out[i,:] = weights[x[i],:]
//   SEQ=8192 rows, EMBED=768 f32 (3072 B/row), VOCAB=50257 (< 65536 -> 16-bit TDM gather indices)
// Pure bandwidth problem (~50 MB total, ~2.2us at 23.3 TB/s). Implemented with the CDNA5
// Tensor Data Mover:
//   TENSOR_LOAD_TO_LDS (gather_mode, 16 rows / 48KB per descriptor)  ->  LDS (double buffered)
//   TENSOR_STORE_FROM_LDS (1-D contiguous 48KB)                      ->  out
// One wave per workgroup drives the DMA; zero VALU work per byte moved.

#define SEQ            8192
#define EMBED          768
#define VOCAB          50257
#define ROWS_PER_CHUNK 16                          // full 16-bit-index gather capacity
#define CHUNKS         (SEQ / ROWS_PER_CHUNK)      // 512
#define NBLOCKS        256
#define CHUNKS_PER_BLK (CHUNKS / NBLOCKS)          // 2

typedef unsigned int v4u __attribute__((ext_vector_type(4)));
typedef int          v8i __attribute__((ext_vector_type(8)));
typedef int          v4i __attribute__((ext_vector_type(4)));

// D# group0 word3: global_addr[56:32] in bits [24:0], type=2 ("image") in bits [31:30]
__device__ static inline unsigned d0_hi(unsigned long long gaddr) {
  return (unsigned)((gaddr >> 32) & 0x01FFFFFFull) | (2u << 30);
}

// Gather-load 16 rows of `EMBED` f32 from wbase, row ids t[0..15], into LDS at ldsaddr.
__device__ static inline void tdm_gather_load(const int* __restrict__ x, int row0,
                                              unsigned long long wbase, unsigned ldsaddr) {
  int t[ROWS_PER_CHUNK];
#pragma unroll
  for (int k = 0; k < ROWS_PER_CHUNK; ++k) t[k] = x[row0 + k];

  v4u g0;
  g0.x = 0x80000001u;                      // count=1, is_restore=0, gather_mode=1, 16-bit idx
  g0.y = ldsaddr;                          // lds_addr (bytes)
  g0.z = (unsigned)(wbase & 0xFFFFFFFFull);// global_addr[31:0]  (tensor base)
  g0.w = d0_hi(wbase);                     // global_addr[56:32] | type=2

  v8i g1;
  g1[0] = (int)(2u << 16);                           // wg_mask=0, data_size=2 (4B), no flags
  g1[1] = (int)((unsigned)EMBED << 16);              // tensor_dim0[15:0]=768 (bits 63:48)
  g1[2] = (int)(((unsigned)VOCAB & 0xFFFFu) << 16);  // tensor_dim1[15:0]   (bits 95:80)
  g1[3] = (int)((unsigned)EMBED << 16);              // tile_dim0=768       (bits 127:112)
  g1[4] = ROWS_PER_CHUNK;                            // tile_dim1 = #valid indices (bits 143:128)
  g1[5] = EMBED;                                     // tensor_dim0_stride[31:0]=768 elements
  g1[6] = 0;                                         // stride hi, tensor_dim1_stride (ignored)
  g1[7] = 0;

  v4i g2, g3;                                        // 16 packed 16-bit row indices
#pragma unroll
  for (int k = 0; k < 4; ++k)
    g2[k] = (t[2 * k + 1] << 16) | (t[2 * k] & 0xFFFF);
#pragma unroll
  for (int k = 0; k < 4; ++k)
    g3[k] = (t[8 + 2 * k + 1] << 16) | (t[8 + 2 * k] & 0xFFFF);

  v8i g23;                                           // groups 2+3 as one 256-bit chunk
#pragma unroll
  for (int k = 0; k < 4; ++k) { g23[k] = g2[k]; g23[4 + k] = g3[k]; }

  __builtin_amdgcn_tensor_load_to_lds(g0, g1, g2, g3, g23, 0);
}

// Store ROWS_PER_CHUNK*EMBED contiguous f32 from LDS at ldsaddr to obase + row0*EMBED*4.
__device__ static inline void tdm_store(unsigned long long obase, int row0, unsigned ldsaddr) {
  const unsigned n = ROWS_PER_CHUNK * EMBED;         // 12288 elements, contiguous
  unsigned long long gaddr = obase + (unsigned long long)row0 * (EMBED * 4u);

  v4u g0;
  g0.x = 0x00000001u;                                // count=1, normal mode
  g0.y = ldsaddr;
  g0.z = (unsigned)(gaddr & 0xFFFFFFFFull);
  g0.w = d0_hi(gaddr);

  v8i g1;
  g1[0] = (int)(2u << 16);                           // data_size = 4B
  g1[1] = (int)((n & 0xFFFFu) << 16);                // tensor_dim0[15:0] = 12288
  g1[2] = (int)(1u << 16);                           // tensor_dim1 = 1
  g1[3] = (int)(n << 16);                            // tile_dim0 = 12288 (fits 16 bits)
  g1[4] = 1;                                         // tile_dim1 = 1, tile_dim2 = 0
  g1[5] = (int)n;                                    // tensor_dim0_stride = 12288
  g1[6] = 0;
  g1[7] = 0;

  v4i gz4 = (v4i){0, 0, 0, 0};
  v8i gz8 = (v8i){0, 0, 0, 0, 0, 0, 0, 0};
  __builtin_amdgcn_tensor_store_from_lds(g0, g1, gz4, gz4, gz8, 0);
}

__global__ __launch_bounds__(32)
void embed_gather_tdm(const int* __restrict__ x, const float* __restrict__ w,
                      float* __restrict__ out) {
  __shared__ float buf[2][ROWS_PER_CHUNK * EMBED];   // 2 x 48 KB double buffer (<< 320 KB WGP LDS)

  const unsigned long long wbase = (unsigned long long)(uintptr_t)w;
  const unsigned long long obase = (unsigned long long)(uintptr_t)out;
  // Flat LDS addresses keep the LDS byte offset in the low 32 bits (ISA 10.2).
  const unsigned lds0 = (unsigned)(uintptr_t)&buf[0][0];
  const unsigned lds1 = (unsigned)(uintptr_t)&buf[1][0];

  const int chunk0 = blockIdx.x * CHUNKS_PER_BLK;

  // Prime the pipeline: load chunk0 into buf0.
  tdm_gather_load(x, chunk0 * ROWS_PER_CHUNK, wbase, lds0);

#pragma unroll
  for (int i = 0; i < CHUNKS_PER_BLK; ++i) {
    const int row_i  = (chunk0 + i) * ROWS_PER_CHUNK;
    const unsigned lds_cur = (i & 1) ? lds1 : lds0;
    if (i + 1 < CHUNKS_PER_BLK) {
      // Prefetch next chunk into the other buffer, then wait until only that
      // load is outstanding (TDM ops from one wave complete in order, so
      // tensorcnt<=1 guarantees previous store + current load are done).
      tdm_gather_load(x, row_i + ROWS_PER_CHUNK, wbase, (i & 1) ? lds0 : lds1);
      __builtin_amdgcn_s_wait_tensorcnt(1);
    } else {
      __builtin_amdgcn_s_wait_tensorcnt(0);
    }
    tdm_store(obase, row_i, lds_cur);
    // Final store drained by S_ENDPGM's implicit wait-idle.
  }
}

extern "C" void kernel_launch(void* const* d_in, const int* in_sizes, int n_in,
                              void* d_out, int out_size, void* d_ws, size_t ws_size,
                              hipStream_t stream) {
  const int*   x = (const int*)d_in[0];     // [1, SEQ] int32 token ids
  const float* w = (const float*)d_in[1];   // [VOCAB, EMBED] f32
  float*     out = (float*)d_out;           // [SEQ, EMBED] f32
  (void)in_sizes; (void)n_in; (void)out_size; (void)d_ws; (void)ws_size;

  embed_gather_tdm<<<NBLOCKS, 32, 0, stream>>>(x, w, out);
}